// CausalGNNCore_68633577390499
// MI455X (gfx1250) — compile-verified
//
#include <hip/hip_runtime.h>

// Problem constants from the reference
#define BB 128
#define DD 64
#define HH 128
#define MM (BB * DD)   // 8192 rows in flattened (b,d) space
#define WSTRIDE 132    // padded LDS column stride (dwords): conflict-free + affine

typedef float v2f __attribute__((ext_vector_type(2)));
typedef float v4f __attribute__((ext_vector_type(4)));
typedef float v8f __attribute__((ext_vector_type(8)));
typedef int   b128_t __attribute__((vector_size(16)));   // matches builtin pointee

#define GLOBAL_AS __attribute__((address_space(1)))
#define LDS_AS    __attribute__((address_space(3)))

// ---------------------------------------------------------------------------
// Async global->LDS staging (CDNA5 GLOBAL_LOAD_ASYNC_TO_LDS_* path), with
// plain LDS store fallbacks if the builtins are absent.
// ---------------------------------------------------------------------------
__device__ __forceinline__ void stage_b128(const float* g, float* l) {
#if __has_builtin(__builtin_amdgcn_global_load_async_to_lds_b128)
  __builtin_amdgcn_global_load_async_to_lds_b128(
      (GLOBAL_AS b128_t*)g, (LDS_AS b128_t*)l, 0, 0);
#else
  *(v4f*)l = *(const v4f*)g;
#endif
}

__device__ __forceinline__ void stage_b32(const float* g, float* l) {
#if __has_builtin(__builtin_amdgcn_global_load_async_to_lds_b32)
  __builtin_amdgcn_global_load_async_to_lds_b32(
      (GLOBAL_AS int*)g, (LDS_AS int*)l, 0, 0);
#else
  *l = *g;
#endif
}

__device__ __forceinline__ void stage_join() {
#if __has_builtin(__builtin_amdgcn_global_load_async_to_lds_b32)
#if __has_builtin(__builtin_amdgcn_s_wait_asynccnt)
  __builtin_amdgcn_s_wait_asynccnt(0);
#else
  asm volatile("s_wait_asynccnt 0x0" ::: "memory");
#endif
#endif
}

// ---------------------------------------------------------------------------
// Kernel 1: h[m,c] = tanh(X[m] * enc_w[c] + enc_b[c])     (elementwise)
// ---------------------------------------------------------------------------
__global__ void enc_kernel(const float* __restrict__ X, const float* __restrict__ ew,
                           const float* __restrict__ eb, float* __restrict__ h) {
  int idx = blockIdx.x * blockDim.x + threadIdx.x;   // 0 .. MM*HH-1
  int m = idx >> 7;           // HH == 128
  int c = idx & (HH - 1);
  h[idx] = tanhf(X[m] * ew[c] + eb[c]);
}

// ---------------------------------------------------------------------------
// Kernel 2: s[i] = sum_j (j!=i) W[j,i]   (tiny)
// ---------------------------------------------------------------------------
__global__ void col_sums_kernel(const float* __restrict__ W, float* __restrict__ s) {
  int i = threadIdx.x;   // 0..DD-1
  float acc = 0.f;
  for (int j = 0; j < DD; ++j)
    if (j != i) acc += W[j * DD + i];
  s[i] = acc;
}

// ---------------------------------------------------------------------------
// Kernel 3 (dominant VALU pass), one block per batch element b:
//   r[b,i,c] = sum_j A[j,i] * relu(u[b,j,c] + v[b,i,c] + msg_b1[c])
// u[b] slice (32KB) and W (16KB) staged in LDS via async b128; the j-loop is
// then pure VALU fed by conflict-free LDS reads (u) and wave-broadcast (W).
// ---------------------------------------------------------------------------
__global__ __launch_bounds__(256)
void fused_msg_agg_kernel(const float* __restrict__ u, const float* __restrict__ v,
                          const float* __restrict__ W, const float* __restrict__ mb1,
                          float* __restrict__ r) {
  __shared__ float u_sh[DD * HH];   // 32 KB
  __shared__ float w_sh[DD * DD];   // 16 KB
  int b = blockIdx.x;
  const float* ub = u + (size_t)b * DD * HH;
  for (int idx = threadIdx.x * 4; idx < DD * HH; idx += 256 * 4)
    stage_b128(ub + idx, &u_sh[idx]);
  for (int idx = threadIdx.x * 4; idx < DD * DD; idx += 256 * 4)
    stage_b128(W + idx, &w_sh[idx]);
  stage_join();
  __syncthreads();

  int c  = threadIdx.x & (HH - 1);  // hidden channel
  int ig = threadIdx.x >> 7;        // 0 or 1: i-interleave group
  float bias = mb1[c];
  for (int i = ig; i < DD; i += 2) {
    float vb = v[((size_t)b * DD + i) * HH + c] + bias;
    float acc = 0.f;
#pragma unroll
    for (int j = 0; j < DD; ++j) {
      float a = (j == i) ? 0.f : w_sh[j * DD + i];
      acc = fmaf(a, fmaxf(u_sh[j * HH + c] + vb, 0.f), acc);
    }
    r[((size_t)b * DD + i) * HH + c] = acc;
  }
}

// ---------------------------------------------------------------------------
// Kernel 4: fp32 WMMA GEMM, K=N=128:
//   out = act( A1 @ W1 [+ A2 @ W2] [+ bias] [+ rowscale[row%D] * bias2] )
// 256 threads = 8 waves; wave owns a 16-row strip across N=128 (8 C tiles).
// Weights live in LDS TRANSPOSED with padded stride 132 dwords (67.6 KB of
// the 320 KB WGP LDS): element (k, col) at dword col*132 + k.
//  - each B fragment {W[kk,col], W[kk+1,col]} is one aligned ds_load_b64
//  - read bank = (4*col + kk) mod 64: all 32 lanes distinct -> conflict-free
//  - address is affine: base(lane) + nt*8448 + ks*16 bytes, all < 64KB, so a
//    single LDS address VGPR + immediate offsets covers all 256 loads
//    (avoids the >256-VGPR / s_set_vgpr_msb churn of a mod-swizzled layout).
// All 32 A fragments are preloaded into VGPRs (LOADcnt) overlapping the async
// LDS staging (ASYNCcnt); steady-state loop is pure ds_load_b64 + WMMA.
// ---------------------------------------------------------------------------
__global__ __launch_bounds__(256)
void wmma_gemm_kernel(const float* __restrict__ A1, const float* __restrict__ W1,
                      const float* __restrict__ A2, const float* __restrict__ W2,
                      const float* __restrict__ bias,
                      const float* __restrict__ rowscale,
                      const float* __restrict__ bias2,
                      float* __restrict__ out, int relu_flag) {
  const int K = HH, N = HH;
  __shared__ float wt[HH * WSTRIDE];   // 67.6 KB transposed weight stage

  int lane = threadIdx.x & 31;
  int wv   = threadIdx.x >> 5;
  int rowTile = (blockIdx.x * 8 + wv) * 16;
  int rsub = lane & 15;
  int koff = (lane >> 4) * 2;                 // lanes 16-31 hold K=2,3 of each step
  int r0   = rowTile + (lane >> 4) * 8;       // C/D: VGPR e -> row r0+e

  v8f acc[8] = {};

  int nphase = A2 ? 2 : 1;
  for (int phase = 0; phase < nphase; ++phase) {
    const float* Ap = phase ? A2 : A1;
    const float* Wp = phase ? W2 : W1;

    __syncthreads();   // wt safe to overwrite (prior phase readers done)
    // Transpose scatter into LDS; global side stays coalesced.
    for (int idx = threadIdx.x; idx < HH * HH; idx += 256) {
      int row = idx >> 7;            // k
      int col = idx & (HH - 1);
      stage_b32(Wp + idx, &wt[col * WSTRIDE + row]);
    }

    // Preload all 32 A fragments while staging is in flight (split counters).
    const float* arow = Ap + (size_t)(rowTile + rsub) * K + koff;
    v2f af[32];
#pragma unroll
    for (int ks = 0; ks < K / 4; ++ks)
      af[ks] = *(const v2f*)(arow + ks * 4);

    stage_join();
    __syncthreads();

#pragma unroll
    for (int ks = 0; ks < K / 4; ++ks) {
#pragma unroll
      for (int nt = 0; nt < 8; ++nt) {
        int colg = nt * 16 + rsub;
        v2f bf = *(const v2f*)&wt[colg * WSTRIDE + ks * 4 + koff];
        acc[nt] = __builtin_amdgcn_wmma_f32_16x16x4_f32(false, af[ks], false, bf,
                                                        (short)0, acc[nt],
                                                        false, false);
      }
    }
  }

#pragma unroll
  for (int nt = 0; nt < 8; ++nt) {
    int colg = nt * 16 + rsub;
    float badd = bias ? bias[colg] : 0.f;
    float b2   = rowscale ? bias2[colg] : 0.f;
#pragma unroll
    for (int e = 0; e < 8; ++e) {
      int rr = r0 + e;
      float cv = acc[nt][e] + badd;
      if (rowscale) cv = fmaf(rowscale[rr & (DD - 1)], b2, cv);
      if (relu_flag) cv = fmaxf(cv, 0.f);
      out[(size_t)rr * N + colg] = cv;
    }
  }
}

// ---------------------------------------------------------------------------
// Kernel 5: out[m] = z[m,:] . dec_w2 + dec_b2   (wave32 GEMV reduction)
// ---------------------------------------------------------------------------
__global__ void gemv_out_kernel(const float* __restrict__ z, const float* __restrict__ w,
                                const float* __restrict__ b, float* __restrict__ out) {
  int m    = blockIdx.x * (blockDim.x >> 5) + (threadIdx.x >> 5);
  int lane = threadIdx.x & 31;
  float acc = 0.f;
  for (int k = lane; k < HH; k += 32)
    acc = fmaf(z[(size_t)m * HH + k], w[k], acc);
  for (int off = 16; off > 0; off >>= 1)
    acc += __shfl_xor(acc, off, 32);
  if (lane == 0) out[m] = acc + b[0];
}

// ---------------------------------------------------------------------------
extern "C" void kernel_launch(void* const* d_in, const int* in_sizes, int n_in,
                              void* d_out, int out_size, void* d_ws, size_t ws_size,
                              hipStream_t stream) {
  (void)in_sizes; (void)n_in; (void)out_size; (void)ws_size;

  const float* X     = (const float*)d_in[0];   // (B,D)
  const float* W     = (const float*)d_in[1];   // (D,D)
  const float* enc_w = (const float*)d_in[2];   // (1,H)
  const float* enc_b = (const float*)d_in[3];   // (H)
  const float* mw1   = (const float*)d_in[4];   // (2H,H)
  const float* mb1   = (const float*)d_in[5];   // (H)
  const float* mw2   = (const float*)d_in[6];   // (H,H)
  const float* mb2   = (const float*)d_in[7];   // (H)
  const float* dw1   = (const float*)d_in[8];   // (2H,H)
  const float* db1   = (const float*)d_in[9];   // (H)
  const float* dw2   = (const float*)d_in[10];  // (H,1)
  const float* db2   = (const float*)d_in[11];  // (1)
  float* out = (float*)d_out;                   // (B,D) fp32

  float* ws = (float*)d_ws;
  const size_t S = (size_t)MM * HH;             // 1M floats per buffer
  float* h   = ws;
  float* u   = ws + 1 * S;
  float* v   = ws + 2 * S;
  float* r   = ws + 3 * S;
  float* agg = ws + 4 * S;
  float* z   = ws + 5 * S;
  float* s   = ws + 6 * S;                      // D floats

  // 1) h = tanh(X ⊗ enc_w + enc_b)
  enc_kernel<<<(MM * HH) / 256, 256, 0, stream>>>(X, enc_w, enc_b, h);
  // 2) u = h @ msg_w1[:H],  v = h @ msg_w1[H:]
  wmma_gemm_kernel<<<MM / 128, 256, 0, stream>>>(h, mw1, nullptr, nullptr,
                                                 nullptr, nullptr, nullptr, u, 0);
  wmma_gemm_kernel<<<MM / 128, 256, 0, stream>>>(h, mw1 + HH * HH, nullptr, nullptr,
                                                 nullptr, nullptr, nullptr, v, 0);
  // 3) s[i] = sum_j!=i W[j,i]
  col_sums_kernel<<<1, DD, 0, stream>>>(W, s);
  // 4) r = sum_j A[j,i] * relu(u + v + msg_b1)   (aggregate BEFORE 2nd msg layer)
  fused_msg_agg_kernel<<<BB, 256, 0, stream>>>(u, v, W, mb1, r);
  // 5) agg = r @ msg_w2 + s[i] * msg_b2
  wmma_gemm_kernel<<<MM / 128, 256, 0, stream>>>(r, mw2, nullptr, nullptr,
                                                 nullptr, s, mb2, agg, 0);
  // 6) z = relu(h @ dec_w1[:H] + agg @ dec_w1[H:] + dec_b1)
  wmma_gemm_kernel<<<MM / 128, 256, 0, stream>>>(h, dw1, agg, dw1 + HH * HH,
                                                 db1, nullptr, nullptr, z, 1);
  // 7) out = z @ dec_w2 + dec_b2
  gemv_out_kernel<<<MM / 8, 256, 0, stream>>>(z, dw2, db2, out);
}